// LengthRegulatorONNX_45904610460078
// MI455X (gfx1250) — compile-verified
//
#include <hip/hip_runtime.h>
#include <stdint.h>

// LengthRegulator: x (B=32, S=512, H=256) f32, duration (B, S) i32 in [0,16)
// out = (B, M=4000, H) f32 gather + mel_len (B,) i32, M = MAX_MEL_LEN = 4000.
//
// Strategy (memory-roofline, ~148 MB total traffic @ 23.3 TB/s ~ 6.4 us):
//   Kernel 1: per-batch inclusive scan of duration into d_ws (async LDS load +
//             LDS Hillis-Steele scan), emit mel_len = min(total, M).
//   Kernel 2: scatter. Block (s, b) holds x[b,s,:] in registers (64 lanes x
//             float4) and streams it to out rows [cum[s-1], cum[s]) with
//             non-temporal b128 stores. Block s==0 also fills the padding
//             rows [total, M) with x[b,0,:] per the reference semantics.

#define S_LEN 512
#define H_DIM 256
#define M_MAX 4000

typedef float v4f __attribute__((ext_vector_type(4)));

__global__ __launch_bounds__(S_LEN)
void lr_scan_kernel(const int* __restrict__ dur,
                    int* __restrict__ cum,
                    int* __restrict__ mel_len) {
  __shared__ int sd[S_LEN];
  const int b = blockIdx.x;
  const int t = threadIdx.x;

  // gfx1250 async global->LDS copy of this batch's durations (ASYNCcnt path).
  // Generic shared-pointer low 32 bits == LDS byte offset on CDNA5.
  const int* gbase = dur + (size_t)b * S_LEN;          // uniform -> SGPR pair
  uint32_t lds_addr = (uint32_t)(uintptr_t)(&sd[t]);   // LDS byte address
  uint32_t voff = (uint32_t)t * 4u;                    // per-lane byte offset
  asm volatile("global_load_async_to_lds_b32 %0, %1, %2"
               :: "v"(lds_addr), "v"(voff), "s"(gbase)
               : "memory");
  asm volatile("s_wait_asynccnt 0" ::: "memory");
  __syncthreads();

  // Hillis-Steele inclusive scan over S_LEN elements in LDS.
  #pragma unroll
  for (int off = 1; off < S_LEN; off <<= 1) {
    int v = (t >= off) ? sd[t - off] : 0;
    __syncthreads();
    sd[t] += v;
    __syncthreads();
  }

  cum[(size_t)b * S_LEN + t] = sd[t];
  if (t == S_LEN - 1) {
    int total = sd[t];
    mel_len[b] = (total < M_MAX) ? total : M_MAX;
  }
}

__global__ __launch_bounds__(64)
void lr_scatter_kernel(const float* __restrict__ x,
                       const int* __restrict__ cum,
                       float* __restrict__ out) {
  const int s = blockIdx.x;       // source frame
  const int b = blockIdx.y;       // batch
  const int lane = threadIdx.x;   // 0..63, lane*4 floats of the H=256 row

  const int* crow = cum + (size_t)b * S_LEN;
  int end   = crow[s];
  int start = (s == 0) ? 0 : crow[s - 1];
  if (end   > M_MAX) end   = M_MAX;
  if (start > M_MAX) start = M_MAX;

  // Load this source row once (coalesced b128), keep in registers.
  const v4f* xrow = (const v4f*)(x + ((size_t)b * S_LEN + s) * H_DIM);
  v4f v = xrow[lane];

  v4f* obase = (v4f*)(out + (size_t)b * M_MAX * H_DIM);

  // Stream to all output rows this frame covers; write-once data -> NT stores
  // so the 131 MB output doesn't evict the reusable x / cumsum from L2.
  for (int m = start; m < end; ++m) {
    __builtin_nontemporal_store(v, &obase[(size_t)m * (H_DIM / 4) + lane]);
  }

  // Reference pads rows [total, M) with x[b, 0, :] (idx forced to 0).
  if (s == 0) {
    int total = crow[S_LEN - 1];
    if (total > M_MAX) total = M_MAX;
    for (int m = total; m < M_MAX; ++m) {
      __builtin_nontemporal_store(v, &obase[(size_t)m * (H_DIM / 4) + lane]);
    }
  }
}

extern "C" void kernel_launch(void* const* d_in, const int* in_sizes, int n_in,
                              void* d_out, int out_size, void* d_ws, size_t ws_size,
                              hipStream_t stream) {
  const float* x  = (const float*)d_in[0];
  const int* dur  = (const int*)d_in[1];
  // d_in[2] is max_len (scalar) — the reference shadows it with MAX_MEL_LEN.

  const int B = in_sizes[1] / S_LEN;  // 32

  float* out     = (float*)d_out;
  int*   mel_len = (int*)(out + (size_t)B * M_MAX * H_DIM);  // second tuple output
  int*   cumws   = (int*)d_ws;                               // B*S ints of scratch

  lr_scan_kernel<<<dim3(B), dim3(S_LEN), 0, stream>>>(dur, cumws, mel_len);
  lr_scatter_kernel<<<dim3(S_LEN, B), dim3(64), 0, stream>>>(x, cumws, out);
}